// GnnLayer_39573828665552
// MI455X (gfx1250) — compile-verified
//
#include <hip/hip_runtime.h>
#include <hip/hip_bf16.h>

// ---------------------------------------------------------------------------
// 2-layer GAT for MI455X (gfx1250, wave32).
//  - GEMMs (x @ W^T) via V_WMMA_F32_16X16X4_F32 (exact fp32 matrix path).
//  - Edge phase: segment softmax + scatter-aggregate with f32 atomics;
//    all intermediates sized to stay resident in the 192MB L2.
// ---------------------------------------------------------------------------

typedef __attribute__((ext_vector_type(2))) float v2f;
typedef __attribute__((ext_vector_type(8))) float v8f;

#define GAT_H   8     // heads per layer
#define GAT_CH  256   // in/out channels per layer
#define GAT_HC  32    // channels per head
#define NEG_SLOPE 0.2f

// ---------------------------------------------------------------------------
// Y[n, o] = sum_k X[n,k] * W[o,k]   (K = O = 256)
// One wave computes one 16x16 tile of Y with 64 chained f32 WMMAs.
// ISA 7.12.2 layouts:
//   A 16x4 : lane L holds row M=(L&15); VGPR0/1 = K = (L>>4)*2 + {0,1}
//   B  4x16: lane L holds col N=(L&15); VGPR0/1 = K = (L>>4)*2 + {0,1}
//   C/D    : lane L holds col N=(L&15); VGPR r  = row M = r + (L>>4)*8
// ---------------------------------------------------------------------------
__global__ void gat_gemm_wmma(const float* __restrict__ X,
                              const float* __restrict__ W,
                              float* __restrict__ Y,
                              int nRowTiles) {
  const int wave = threadIdx.x >> 5;          // 8 waves / block
  const int lane = threadIdx.x & 31;
  const int tile = blockIdx.x * 8 + wave;
  if (tile >= nRowTiles * (GAT_CH / 16)) return;   // wave-uniform: EXEC stays all-1s

  const int rowTile = tile >> 4;              // over N
  const int colTile = tile & 15;              // over 256 output channels
  const int l15   = lane & 15;
  const int lhalf = lane >> 4;                // 0 or 1

  const v2f* Xp = (const v2f*)(X + (size_t)(rowTile * 16 + l15) * GAT_CH);
  const v2f* Wp = (const v2f*)(W + (size_t)(colTile * 16 + l15) * GAT_CH);

  v8f acc = {};
#pragma unroll 16
  for (int s = 0; s < 64; ++s) {              // K = 256 = 64 * 4
    v2f a = Xp[s * 2 + lhalf];                // X[row, 4s + 2*lhalf + {0,1}]
    v2f b = Wp[s * 2 + lhalf];                // W[col, 4s + 2*lhalf + {0,1}]
    acc = __builtin_amdgcn_wmma_f32_16x16x4_f32(
        /*neg_a=*/false, a, /*neg_b=*/false, b,
        /*c_mod=*/(short)0, acc, /*reuse_a=*/false, /*reuse_b=*/false);
  }

  float* Yt = Y + (size_t)(rowTile * 16 + lhalf * 8) * GAT_CH + colTile * 16 + l15;
#pragma unroll
  for (int r = 0; r < 8; ++r)
    Yt[(size_t)r * GAT_CH] = acc[r];
}

// Scalar tail for rows not covered by 16-row tiles (N=50000 is divisible, so
// this normally launches zero blocks; kept for shape robustness).
__global__ void gat_gemm_tail(const float* __restrict__ X,
                              const float* __restrict__ W,
                              float* __restrict__ Y,
                              int rowStart, int nRows) {
  int idx = blockIdx.x * blockDim.x + threadIdx.x;
  int row = rowStart + (idx >> 8);
  int o   = idx & 255;
  if (row >= nRows) return;
  const float* x = X + (size_t)row * GAT_CH;
  const float* w = W + (size_t)o * GAT_CH;
  float s = 0.f;
#pragma unroll 8
  for (int k = 0; k < GAT_CH; ++k) s += x[k] * w[k];
  Y[(size_t)row * GAT_CH + o] = s;
}

// a_src[n,h] = dot(xw[n,h,:], att_src[h,:]);  a_dst likewise.
__global__ void gat_attn_scores(const float* __restrict__ xw,
                                const float* __restrict__ att_src,
                                const float* __restrict__ att_dst,
                                float* __restrict__ a_src,
                                float* __restrict__ a_dst, int n) {
  int idx = blockIdx.x * blockDim.x + threadIdx.x;   // n * H threads
  if (idx >= n * GAT_H) return;
  int node = idx >> 3, h = idx & 7;
  const float* row = xw + (size_t)node * GAT_CH + h * GAT_HC;
  const float* as  = att_src + h * GAT_HC;
  const float* ad  = att_dst + h * GAT_HC;
  float s = 0.f, d = 0.f;
#pragma unroll
  for (int c = 0; c < GAT_HC; ++c) { float v = row[c]; s += v * as[c]; d += v * ad[c]; }
  a_src[idx] = s;
  a_dst[idx] = d;
}

__global__ void gat_fill_f32(float* __restrict__ p, float v, int n) {
  int i = blockIdx.x * blockDim.x + threadIdx.x;
  if (i < n) p[i] = v;
}

// float atomic max via ordered-int trick (init must be -inf)
__device__ __forceinline__ void atomicMaxF32(float* addr, float val) {
  if (val >= 0.f) atomicMax((int*)addr, __float_as_int(val));
  else            atomicMin((unsigned int*)addr, __float_as_uint(val));
}

__device__ __forceinline__ float leaky(float v) {
  return v >= 0.f ? v : NEG_SLOPE * v;
}

// Pass 1: segment max of leaky-relu(e) grouped by dst.
__global__ void gat_edge_max(const int* __restrict__ eidx, int E,
                             const float* __restrict__ a_src,
                             const float* __restrict__ a_dst,
                             float* __restrict__ m) {
  int idx = blockIdx.x * blockDim.x + threadIdx.x;   // E * H threads
  if (idx >= E * GAT_H) return;
  int e = idx >> 3, h = idx & 7;
  int s = eidx[e], d = eidx[E + e];
  float v = leaky(a_src[s * GAT_H + h] + a_dst[d * GAT_H + h]);
  atomicMaxF32(&m[d * GAT_H + h], v);
}

// Pass 2: denom[d,h] += exp(e - m[d,h])
__global__ void gat_edge_expsum(const int* __restrict__ eidx, int E,
                                const float* __restrict__ a_src,
                                const float* __restrict__ a_dst,
                                const float* __restrict__ m,
                                float* __restrict__ denom) {
  int idx = blockIdx.x * blockDim.x + threadIdx.x;
  if (idx >= E * GAT_H) return;
  int e = idx >> 3, h = idx & 7;
  int s = eidx[e], d = eidx[E + e];
  int dh = d * GAT_H + h;
  float v = leaky(a_src[s * GAT_H + h] + a_dst[dh]);
  atomicAdd(&denom[dh], expf(v - m[dh]));
}

// Pass 3: out[d, h*32+c] += alpha * xw[s, h*32+c]
__global__ void gat_edge_aggregate(const int* __restrict__ eidx, int E,
                                   const float* __restrict__ a_src,
                                   const float* __restrict__ a_dst,
                                   const float* __restrict__ m,
                                   const float* __restrict__ denom,
                                   const float* __restrict__ xw,
                                   float* __restrict__ out) {
  int idx = blockIdx.x * blockDim.x + threadIdx.x;
  if (idx >= E * GAT_H) return;
  int e = idx >> 3, h = idx & 7;
  int s = eidx[e], d = eidx[E + e];
  int dh = d * GAT_H + h;
  float v = leaky(a_src[s * GAT_H + h] + a_dst[dh]);
  float alpha = expf(v - m[dh]) / (denom[dh] + 1e-16f);
  const float* srcRow = xw  + (size_t)s * GAT_CH + h * GAT_HC;
  float*       dstRow = out + (size_t)d * GAT_CH + h * GAT_HC;
#pragma unroll
  for (int c = 0; c < GAT_HC; ++c)
    atomicAdd(&dstRow[c], srcRow[c] * alpha);
}

// h = gelu(out + b)  (exact gelu, matches jax.nn.gelu approximate=False)
__global__ void gat_bias_gelu(float* __restrict__ h,
                              const float* __restrict__ bias, int total) {
  int i = blockIdx.x * blockDim.x + threadIdx.x;
  if (i >= total) return;
  float x = h[i] + bias[i & (GAT_CH - 1)];
  h[i] = 0.5f * x * (1.f + erff(x * 0.70710678118654752f));
}

// d_out[o] = out2[(N-1)*256 + o] + b2[o]
__global__ void gat_final_row(const float* __restrict__ out2,
                              const float* __restrict__ b2,
                              float* __restrict__ dOut, int n) {
  int o = threadIdx.x;
  dOut[o] = out2[(size_t)(n - 1) * GAT_CH + o] + b2[o];
}

// ---------------------------------------------------------------------------

static void run_gat_layer(const float* X, const float* W,
                          const float* att_src, const float* att_dst,
                          const int* eidx, int n, int E,
                          float* xw, float* out, float* a_s, float* a_d,
                          float* m, float* den, hipStream_t stream) {
  // GEMM: xw = X @ W^T
  int rowTiles = n / 16;
  int tiles = rowTiles * (GAT_CH / 16);
  if (tiles > 0)
    gat_gemm_wmma<<<(tiles + 7) / 8, 256, 0, stream>>>(X, W, xw, rowTiles);
  int remRows = n - rowTiles * 16;
  if (remRows > 0)
    gat_gemm_tail<<<(remRows * GAT_CH + 255) / 256, 256, 0, stream>>>(
        X, W, xw, rowTiles * 16, n);

  // attention scores
  int nh = n * GAT_H;
  gat_attn_scores<<<(nh + 255) / 256, 256, 0, stream>>>(xw, att_src, att_dst,
                                                        a_s, a_d, n);
  // init segment buffers
  hipMemsetAsync(out, 0, (size_t)n * GAT_CH * sizeof(float), stream);
  hipMemsetAsync(den, 0, (size_t)nh * sizeof(float), stream);
  gat_fill_f32<<<(nh + 255) / 256, 256, 0, stream>>>(m, -INFINITY, nh);

  // edge phase
  int eh = E * GAT_H;
  int eb = (eh + 255) / 256;
  gat_edge_max<<<eb, 256, 0, stream>>>(eidx, E, a_s, a_d, m);
  gat_edge_expsum<<<eb, 256, 0, stream>>>(eidx, E, a_s, a_d, m, den);
  gat_edge_aggregate<<<eb, 256, 0, stream>>>(eidx, E, a_s, a_d, m, den, xw, out);
}

extern "C" void kernel_launch(void* const* d_in, const int* in_sizes, int n_in,
                              void* d_out, int out_size, void* d_ws, size_t ws_size,
                              hipStream_t stream) {
  const float* x        = (const float*)d_in[0];
  const int*   eidx     = (const int*)  d_in[1];
  const float* W1       = (const float*)d_in[2];
  const float* att_src1 = (const float*)d_in[3];
  const float* att_dst1 = (const float*)d_in[4];
  const float* b1       = (const float*)d_in[5];
  const float* W2       = (const float*)d_in[6];
  const float* att_src2 = (const float*)d_in[7];
  const float* att_dst2 = (const float*)d_in[8];
  const float* b2       = (const float*)d_in[9];
  float* out = (float*)d_out;

  const int N = in_sizes[0] / GAT_CH;   // 50000
  const int E = in_sizes[1] / 2;        // 800000

  // workspace layout (floats): xw | hbuf | a_src | a_dst | m | denom
  float* ws   = (float*)d_ws;
  float* xw   = ws;                                  // N*256 (xw1, then xw2)
  float* hbuf = xw   + (size_t)N * GAT_CH;           // N*256 (out1/h1, then out2)
  float* a_s  = hbuf + (size_t)N * GAT_CH;           // N*8
  float* a_d  = a_s  + (size_t)N * GAT_H;            // N*8
  float* m    = a_d  + (size_t)N * GAT_H;            // N*8
  float* den  = m    + (size_t)N * GAT_H;            // N*8
  (void)ws_size; (void)n_in; (void)out_size;

  // ----- layer 1: x -> hbuf -----
  run_gat_layer(x, W1, att_src1, att_dst1, eidx, N, E,
                xw, hbuf, a_s, a_d, m, den, stream);
  // h1 = gelu(out1 + b1), in place
  int total = N * GAT_CH;
  gat_bias_gelu<<<(total + 255) / 256, 256, 0, stream>>>(hbuf, b1, total);

  // ----- layer 2: hbuf -> xw (gemm) -> hbuf (aggregate, reused) -----
  run_gat_layer(hbuf, W2, att_src2, att_dst2, eidx, N, E,
                xw, hbuf, a_s, a_d, m, den, stream);
  // NOTE: run_gat_layer zeroes `out` (hbuf) only AFTER the GEMM that reads
  // hbuf as X has been enqueued; stream ordering keeps this correct.

  // final: single row N-1 plus bias
  gat_final_row<<<1, GAT_CH, 0, stream>>>(hbuf, b2, out, N);
}